// skip_gram_with_Hierarchy_43808666419218
// MI455X (gfx1250) — compile-verified
//
#include <hip/hip_runtime.h>

// Problem sizes (from reference)
#define NS     4096
#define PD     170      // P*D = 10*17
#define DIM    300
#define TILE   16
#define NTILES 11       // ceil(170/16)
#define CHUNKS (DIM / 4)        // 75 float4 chunks per row
#define OPS_PER_TILE 48         // 16 rows * 3 async b128 issues

typedef __attribute__((ext_vector_type(2))) float v2f;
typedef __attribute__((ext_vector_type(8))) float v8f;
typedef __attribute__((ext_vector_type(4))) int   v4i;
typedef __attribute__((address_space(1))) v4i gv4i;   // global v4i
typedef __attribute__((address_space(3))) v4i lv4i;   // LDS v4i

__shared__ __align__(16) float proj_s[2][DIM];
__shared__ __align__(16) float hv_s[2][2][TILE][DIM];   // [wave][buf][row][dim]

// Issue one tile's 16 gathered emb2 rows as async global->LDS b128 copies.
__device__ __forceinline__ void stage_tile_async(
    const float* __restrict__ emb2, const int* __restrict__ paths,
    int n, int tile, float (*buf)[DIM], int lane)
{
    #pragma unroll
    for (int r = 0; r < TILE; ++r) {
        int j   = tile * TILE + r;
        int row = paths[(long)n * PD + (j < PD ? j : PD - 1)];
        const float4* g = (const float4*)(emb2 + (long)row * DIM);
        float4*       s = (float4*)(&buf[r][0]);
        #pragma unroll
        for (int c = 0; c < 3; ++c) {
            int idx = lane + c * 32;
            if (idx < CHUNKS) {
                __builtin_amdgcn_global_load_async_to_lds_b128(
                    (gv4i*)(g + idx), (lv4i*)(s + idx), 0, 0);
            }
        }
    }
}

__global__ __launch_bounds__(64) void hsoftmax_wmma_kernel(
    const int*   __restrict__ word_idx,   // [NS]
    const int*   __restrict__ paths,      // [NS, PD]
    const int*   __restrict__ labels,     // [NS, PD]
    const float* __restrict__ emb1,       // [VOCAB, DIM]
    const float* __restrict__ emb2,       // [VOCAB-1, DIM]
    float*       __restrict__ out)        // [2, NS, PD] (output, target)
{
    const int wave = threadIdx.x >> 5;
    const int lane = threadIdx.x & 31;
    const int n    = blockIdx.x * 2 + wave;         // one sample per wave

    // Stage proj = emb1[word_idx[n]] (coalesced); prefetch tile 0 async.
    const long prow = (long)word_idx[n] * DIM;
    for (int i = lane; i < DIM; i += 32)
        proj_s[wave][i] = emb1[prow + i];

    stage_tile_async(emb2, paths, n, 0, hv_s[wave][0], lane);

    __syncthreads();   // proj_s visible to all lanes (once, uniform)

    const int  half   = lane >> 4;      // 0: lanes 0-15 (K=0,1), 1: lanes 16-31 (K=2,3)
    const int  l      = lane & 15;      // A-matrix row (node within tile)
    const int  sel    = lane & 7;       // which acc VGPR this lane extracts
    const bool storer = (lane & 8) == 0;

    for (int tile = 0; tile < NTILES; ++tile) {
        const int cur = tile & 1;

        // Prefetch next tile into the other buffer, then wait for current tile.
        // Async loads complete in order, so cnt<=OPS_PER_TILE => tile `tile` done.
        if (tile + 1 < NTILES) {
            stage_tile_async(emb2, paths, n, tile + 1, hv_s[wave][cur ^ 1], lane);
            __builtin_amdgcn_s_wait_asynccnt(OPS_PER_TILE);
        } else {
            __builtin_amdgcn_s_wait_asynccnt(0);
        }
        __builtin_amdgcn_wave_barrier();   // keep ds reads below the wait

        // ---- 16 dot products via accumulating V_WMMA_F32_16X16X4_F32.
        // A = hv rows (16x4), B = proj chunk broadcast into all 16 columns.
        v8f acc = {};
        #pragma unroll 5
        for (int kk = 0; kk < DIM; kk += 4) {
            int kb = kk + half * 2;
            v2f a = *(const v2f*)(&hv_s[wave][cur][l][kb]);  // conflict-free ds_load_b64
            v2f b = *(const v2f*)(&proj_s[wave][kb]);        // broadcast ds_load_b64
            acc = __builtin_amdgcn_wmma_f32_16x16x4_f32(
                false, a, false, b, (short)0, acc, false, false);
        }

        // ---- Extract: node m<8 -> acc[m] (lanes 0-15); m>=8 -> acc[m-8] (lanes 16-31).
        float v = acc[0];
        v = (sel == 1) ? acc[1] : v;
        v = (sel == 2) ? acc[2] : v;
        v = (sel == 3) ? acc[3] : v;
        v = (sel == 4) ? acc[4] : v;
        v = (sel == 5) ? acc[5] : v;
        v = (sel == 6) ? acc[6] : v;
        v = (sel == 7) ? acc[7] : v;

        int j = tile * TILE + half * 8 + sel;
        if (storer && j < PD) {
            float o   = 1.0f / (1.0f + expf(-v));         // sigmoid
            int   msk = (o >= 0.5f) ? 1 : 0;
            int   lab = labels[(long)n * PD + j];
            out[(long)n * PD + j]                 = o;                          // output
            out[(long)NS * PD + (long)n * PD + j] = (msk == lab) ? 1.0f : 0.0f; // target
        }
    }
}

extern "C" void kernel_launch(void* const* d_in, const int* in_sizes, int n_in,
                              void* d_out, int out_size, void* d_ws, size_t ws_size,
                              hipStream_t stream) {
    const int*   word_idx = (const int*)  d_in[0];
    const int*   paths    = (const int*)  d_in[1];
    const int*   labels   = (const int*)  d_in[2];
    const float* emb1     = (const float*)d_in[3];
    const float* emb2     = (const float*)d_in[4];
    float*       out      = (float*)      d_out;

    (void)in_sizes; (void)n_in; (void)out_size; (void)d_ws; (void)ws_size;

    hsoftmax_wmma_kernel<<<NS / 2, 64, 0, stream>>>(
        word_idx, paths, labels, emb1, emb2, out);
}